// FinancialFeatureCoupling_60550448939399
// MI455X (gfx1250) — compile-verified
//
#include <hip/hip_runtime.h>

typedef float v2f __attribute__((ext_vector_type(2)));
typedef float v8f __attribute__((ext_vector_type(8)));

#define NPAIRS 6
#define HID 16
#define LN_EPS 1e-5f

__global__ __launch_bounds__(256) void ffc_kernel(
    const float* __restrict__ x,  const float* __restrict__ W1,
    const float* __restrict__ b1, const float* __restrict__ g1,
    const float* __restrict__ be1,const float* __restrict__ W2,
    const float* __restrict__ b2, const float* __restrict__ Wf,
    const float* __restrict__ bf, const float* __restrict__ gf,
    const float* __restrict__ bef, float* __restrict__ out, int ntiles)
{
    const int  lane = threadIdx.x & 31;
    const int  l16  = lane & 15;
    const bool hi   = lane >= 16;
    const int  off8 = hi ? 8 : 0;      // hi half handles hid 8..15 (M = r+8)
    const int  wib  = threadIdx.x >> 5;
    const int  waveId = blockIdx.x * (blockDim.x >> 5) + wib;
    const int  nwaves = gridDim.x * (blockDim.x >> 5);

    const int pi_[NPAIRS] = {0, 0, 0, 1, 1, 2};
    const int pj_[NPAIRS] = {1, 2, 3, 2, 3, 3};

    // ---- A operand (loop-invariant): A = W1_p^T, M=hid(lane 0-15), K0/K1 rows.
    // Hi half supplies the zero-padded K2/K3 rows.
    float a0[NPAIRS], a1[NPAIRS];
#pragma unroll
    for (int p = 0; p < NPAIRS; ++p) {
        a0[p] = hi ? 0.0f : W1[p * 32 + l16];        // W1[p][0][h]
        a1[p] = hi ? 0.0f : W1[p * 32 + 16 + l16];   // W1[p][1][h]
    }

    for (int tile = waveId; tile < ntiles; tile += nwaves) {
        const int row0 = tile << 4;
        // Lane handles row l16 (hi half duplicates the load -> same cacheline).
        const float4 xr = *reinterpret_cast<const float4*>(x + (size_t)(row0 + l16) * 4);
        const float xa[4] = {xr.x, xr.y, xr.z, xr.w};

        float po[NPAIRS];
#pragma unroll
        for (int p = 0; p < NPAIRS; ++p) {
            // B = X^T (4 x 16): lanes 0-15 carry K=0/K=1 (x_i, x_j); hi half = K=2/3 = 0.
            v2f a;  a.x = a0[p];               a.y = a1[p];
            v2f bm; bm.x = hi ? 0.0f : xa[pi_[p]];
                    bm.y = hi ? 0.0f : xa[pj_[p]];

            // C[m][n] = b1[p][m]: per-lane-half 8 consecutive floats of b1.
            const float4 c0 = *reinterpret_cast<const float4*>(b1 + p * HID + off8);
            const float4 c1 = *reinterpret_cast<const float4*>(b1 + p * HID + off8 + 4);
            v8f c; c[0]=c0.x; c[1]=c0.y; c[2]=c0.z; c[3]=c0.w;
                   c[4]=c1.x; c[5]=c1.y; c[6]=c1.z; c[7]=c1.w;

            // D[M=hid, N=row] = W1_p^T * X_p^T + b1  (exact fp32)
            v8f d = __builtin_amdgcn_wmma_f32_16x16x4_f32(
                false, a, false, bm, (short)0, c, false, false);

            // ---- LayerNorm over hid: 8 values in this lane + 8 in partner (lane^16)
            float ps = 0.0f, pq = 0.0f;
#pragma unroll
            for (int r = 0; r < 8; ++r) {
                const float v = d[r];
                ps += v;
                pq = fmaf(v, v, pq);
            }
            ps += __shfl_xor(ps, 16, 32);   // combine the two hid halves
            pq += __shfl_xor(pq, 16, 32);
            const float mu  = ps * (1.0f / 16.0f);
            const float var = pq * (1.0f / 16.0f) - mu * mu;
            const float inv = rsqrtf(var + LN_EPS);

            // ---- affine + ReLU + Linear(16->1), params indexed by hid = r + off8
            const float4 g0  = *reinterpret_cast<const float4*>(g1  + p * HID + off8);
            const float4 g1v = *reinterpret_cast<const float4*>(g1  + p * HID + off8 + 4);
            const float4 e0  = *reinterpret_cast<const float4*>(be1 + p * HID + off8);
            const float4 e1v = *reinterpret_cast<const float4*>(be1 + p * HID + off8 + 4);
            const float4 w0  = *reinterpret_cast<const float4*>(W2  + p * HID + off8);
            const float4 w1v = *reinterpret_cast<const float4*>(W2  + p * HID + off8 + 4);
            const float ga[8] = {g0.x,g0.y,g0.z,g0.w,g1v.x,g1v.y,g1v.z,g1v.w};
            const float ea[8] = {e0.x,e0.y,e0.z,e0.w,e1v.x,e1v.y,e1v.z,e1v.w};
            const float wa[8] = {w0.x,w0.y,w0.z,w0.w,w1v.x,w1v.y,w1v.z,w1v.w};

            float acc = 0.0f;
#pragma unroll
            for (int r = 0; r < 8; ++r) {
                float hn = (d[r] - mu) * inv * ga[r] + ea[r];
                hn = fmaxf(hn, 0.0f);
                acc = fmaf(hn, wa[r], acc);
            }
            acc += __shfl_xor(acc, 16, 32); // full 16-element dot product
            po[p] = acc + b2[p];            // pair_out for row l16 (replicated)
        }

        // ---- fusion: Linear(10->4) -> LN(4) -> tanh, all in registers, row = l16
        if (!hi) {
            float comb[10];
            comb[0]=xa[0]; comb[1]=xa[1]; comb[2]=xa[2]; comb[3]=xa[3];
#pragma unroll
            for (int p = 0; p < NPAIRS; ++p) comb[4 + p] = po[p];

            float f[4];
#pragma unroll
            for (int d = 0; d < 4; ++d) {
                float acc = bf[d];
#pragma unroll
                for (int cdx = 0; cdx < 10; ++cdx) acc = fmaf(comb[cdx], Wf[cdx * 4 + d], acc);
                f[d] = acc;
            }
            const float mu = 0.25f * (f[0] + f[1] + f[2] + f[3]);
            float var = 0.0f;
#pragma unroll
            for (int d = 0; d < 4; ++d) { const float dv = f[d] - mu; var += dv * dv; }
            var *= 0.25f;
            const float inv = rsqrtf(var + LN_EPS);

            float4 o;
            o.x = tanhf((f[0] - mu) * inv * gf[0] + bef[0]);
            o.y = tanhf((f[1] - mu) * inv * gf[1] + bef[1]);
            o.z = tanhf((f[2] - mu) * inv * gf[2] + bef[2]);
            o.w = tanhf((f[3] - mu) * inv * gf[3] + bef[3]);
            *reinterpret_cast<float4*>(out + (size_t)(row0 + l16) * 4) = o;
        }
    }
}

extern "C" void kernel_launch(void* const* d_in, const int* in_sizes, int n_in,
                              void* d_out, int out_size, void* d_ws, size_t ws_size,
                              hipStream_t stream) {
    const float* x   = (const float*)d_in[0];
    const float* W1  = (const float*)d_in[1];
    const float* b1  = (const float*)d_in[2];
    const float* g1  = (const float*)d_in[3];
    const float* be1 = (const float*)d_in[4];
    const float* W2  = (const float*)d_in[5];
    const float* b2  = (const float*)d_in[6];
    const float* Wf  = (const float*)d_in[7];
    const float* bf  = (const float*)d_in[8];
    const float* gf  = (const float*)d_in[9];
    const float* bef = (const float*)d_in[10];
    float* out = (float*)d_out;

    const int nrows  = in_sizes[0] / 4;   // 524288
    const int ntiles = nrows / 16;        // 16 rows per wave tile

    int blocks = (ntiles + 7) / 8;        // 8 waves per 256-thread block
    if (blocks > 2048) blocks = 2048;     // grid-stride: >=2 tiles/wave amortizes hoisting
    if (blocks < 1) blocks = 1;

    ffc_kernel<<<blocks, 256, 0, stream>>>(x, W1, b1, g1, be1, W2, b2, Wf, bf, gf, bef,
                                           out, ntiles);
}